// DENlayer_34059090657517
// MI455X (gfx1250) — compile-verified
//
#include <hip/hip_runtime.h>

// ---------------------------------------------------------------------------
// Mahalanobis mixture distances:  out[b,e] = (x_b - c_e) Sigma_e^{-1} (x_b - c_e)^T
//   E=32, B=8192, D=256, fp32.  ~34.4 GFLOP GEMM-dominated -> fp32 WMMA.
// Phase 1: Gauss-Jordan inversion of the 32 SPD Sigmas into d_ws (8 MB).
// Phase 2: v_wmma_f32_16x16x4_f32 tiled dif @ inv + elementwise row dot, with
//          DOUBLE-BUFFERED async global->LDS panel staging. The two panel
//          buffers are distinct __shared__ arrays + a 2x-unrolled pipeline so
//          all DS addresses are constant-base (ds_load_2addr offset folding).
// ---------------------------------------------------------------------------

typedef __attribute__((ext_vector_type(2))) float v2f;
typedef __attribute__((ext_vector_type(8))) float v8f;

constexpr int kE = 32;
constexpr int kB = 8192;
constexpr int kD = 256;

constexpr int kRowsPerWave = 16;
constexpr int kWaves      = 4;                       // 128 threads / block
constexpr int kRowsPerBlk = kRowsPerWave * kWaves;   // 64
constexpr int kDifLd      = 260;                     // 260 % 64 == 4 -> conflict-free
constexpr int kPanels     = kD / 16;                 // 16

// ---------------------------------------------------------------------------
// CDNA5 async global->LDS copy (ASYNCcnt-tracked), via inline asm.
// ---------------------------------------------------------------------------
__device__ __forceinline__ void async_copy_b128(unsigned lds_off, const void* g) {
  asm volatile("global_load_async_to_lds_b128 %0, %1, off"
               :: "v"(lds_off), "v"((unsigned long long)(uintptr_t)g)
               : "memory");
}
__device__ __forceinline__ void wait_async0() {
  asm volatile("s_wait_asynccnt 0x0" ::: "memory");
}
__device__ __forceinline__ unsigned lds_offset(const void* p) {
  // generic LDS pointer = {shared aperture, offset32}; low 32 bits = LDS addr
  return (unsigned)(uintptr_t)p;
}

// ---------------------------------------------------------------------------
// Kernel 1: per-expert in-place Gauss-Jordan inversion (SPD, no pivoting).
// ---------------------------------------------------------------------------
__global__ __launch_bounds__(256) void invert_sigma_kernel(
    const float* __restrict__ sigmas,   // [E, D, D]
    float* __restrict__ inv)            // [E, D, D] workspace
{
  __shared__ float s_row[kD];

  const int e   = blockIdx.x;
  const int tid = threadIdx.x;          // 0..255 == row/col index
  const float* S = sigmas + (size_t)e * kD * kD;
  float*       M = inv    + (size_t)e * kD * kD;

  for (int i = tid; i < kD * kD / 4; i += 256)
    ((float4*)M)[i] = ((const float4*)S)[i];
  __threadfence_block();
  __syncthreads();

  for (int k = 0; k < kD; ++k) {
    const float invp = 1.0f / M[(size_t)k * kD + k];
    const float oldv = M[(size_t)k * kD + tid];
    __syncthreads();

    const float rv = (tid == k) ? invp : oldv * invp;
    s_row[tid] = rv;
    M[(size_t)k * kD + tid] = rv;
    __threadfence_block();
    __syncthreads();

    if (tid != k) {
      float* row = M + (size_t)tid * kD;
      const float f   = row[k];
      const float fik = -f * s_row[k];
      for (int j = 0; j < kD; ++j) {
        const float nv = row[j] - f * s_row[j];
        row[j] = (j == k) ? fik : nv;
      }
    }
    __threadfence_block();
    __syncthreads();
  }
}

// ---------------------------------------------------------------------------
// Helpers for kernel 2 (force-inlined so LDS bases stay compile-time consts).
// ---------------------------------------------------------------------------

// Issue async loads for one 256x16 panel of inv into an LDS buffer.
__device__ __forceinline__ void stage_panel_async(const float* __restrict__ invE,
                                                  int kp, unsigned pbase, int tid) {
  const float* src = invE + kp * 16;
  #pragma unroll
  for (int i = 0; i < 8; ++i) {               // 1024 x 16B chunks / 128 threads
    const int c = i * 128 + tid;
    const int r = c >> 2, q = c & 3;          // panel row, 16B chunk in row
    async_copy_b128(pbase + (unsigned)(r * 16 + q * 4) * 4u,
                    src + (size_t)r * kD + q * 4);
  }
}

// T(16x16) = dif(16x256) @ panel(256x16) via 64 WMMAs, then accumulate
// mah[v] += T[v] * dif[m, kp*16+n]   (C layout: lanes 0-15 M=v, 16-31 M=v+8).
__device__ __forceinline__ void panel_compute(const float* __restrict__ panel,
                                              const float* __restrict__ difW,
                                              int kp, int n, int hf, int koff,
                                              float mah[8]) {
  v8f c0 = {0.f,0.f,0.f,0.f,0.f,0.f,0.f,0.f};
  v8f c1 = {0.f,0.f,0.f,0.f,0.f,0.f,0.f,0.f};
  #pragma unroll 8
  for (int d0 = 0; d0 < kD; d0 += 8) {
    {
      const int dk = d0 + koff;
      v2f a; a.x = difW[n * kDifLd + dk]; a.y = difW[n * kDifLd + dk + 1];
      v2f b; b.x = panel[dk * 16 + n];    b.y = panel[(dk + 1) * 16 + n];
      c0 = __builtin_amdgcn_wmma_f32_16x16x4_f32(false, a, false, b,
                                                 (short)0, c0, false, false);
    }
    {
      const int dk = d0 + 4 + koff;
      v2f a; a.x = difW[n * kDifLd + dk]; a.y = difW[n * kDifLd + dk + 1];
      v2f b; b.x = panel[dk * 16 + n];    b.y = panel[(dk + 1) * 16 + n];
      c1 = __builtin_amdgcn_wmma_f32_16x16x4_f32(false, a, false, b,
                                                 (short)0, c1, false, false);
    }
  }
  const v8f t = c0 + c1;
  #pragma unroll
  for (int v = 0; v < 8; ++v) {
    const int m = v + 8 * hf;
    mah[v] += t[v] * difW[m * kDifLd + kp * 16 + n];
  }
}

// ---------------------------------------------------------------------------
// Kernel 2: mah[b,e] via fp32 WMMA + double-buffered async panel staging.
//   grid = (B/64, E), block = 128 threads (4 wave32s), each wave 16 rows.
// ---------------------------------------------------------------------------
__global__ __launch_bounds__(128) void mahal_wmma_kernel(
    const float* __restrict__ x,      // [B, D]
    const float* __restrict__ cent,   // [E, 1, D]
    const float* __restrict__ inv,    // [E, D, D]
    float* __restrict__ out)          // [B, E]
{
  __shared__ float s_dif[kWaves][kRowsPerWave * kDifLd]; // 4*16*260*4B = 65 KB
  __shared__ float s_panelA[kD * 16];                    // 16 KB, buffer A
  __shared__ float s_panelB[kD * 16];                    // 16 KB, buffer B

  const int e    = blockIdx.y;
  const int tid  = threadIdx.x;       // 0..127
  const int wave = tid >> 5;
  const int lane = tid & 31;
  const int n    = lane & 15;
  const int hf   = lane >> 4;
  const int koff = 2 * hf;
  const int b0   = blockIdx.x * kRowsPerBlk + wave * kRowsPerWave;

  const float* invE = inv + (size_t)e * kD * kD;
  const unsigned pbaseA = lds_offset(&s_panelA[0]);
  const unsigned pbaseB = lds_offset(&s_panelB[0]);

  // ---- kick off async staging of panel 0 -> A (overlaps dif staging below)
  stage_panel_async(invE, 0, pbaseA, tid);

  // ---- stage dif = x - centroid for this wave's 16 rows (float4, coalesced)
  {
    const float* cb = cent + (size_t)e * kD;
    for (int i = lane; i < kRowsPerWave * (kD / 4); i += 32) {
      const int m  = i / (kD / 4);
      const int c4 = (i % (kD / 4)) * 4;
      const float4 xv = *(const float4*)(x + (size_t)(b0 + m) * kD + c4);
      const float4 cv = *(const float4*)(cb + c4);
      *(float4*)&s_dif[wave][m * kDifLd + c4] =
          make_float4(xv.x - cv.x, xv.y - cv.y, xv.z - cv.z, xv.w - cv.w);
    }
  }
  wait_async0();
  __syncthreads();

  const float* difW = s_dif[wave];
  float mah[8];
  #pragma unroll
  for (int v = 0; v < 8; ++v) mah[v] = 0.0f;

  // ---- software pipeline, 2x unrolled so each half uses a constant LDS base
  for (int kp = 0; kp < kPanels; kp += 2) {
    // prefetch kp+1 -> B (B's last readers drained by previous barrier)
    stage_panel_async(invE, kp + 1, pbaseB, tid);
    panel_compute(s_panelA, difW, kp, n, hf, koff, mah);
    wait_async0();
    __syncthreads();

    // prefetch kp+2 -> A (A's readers just drained)
    if (kp + 2 < kPanels)
      stage_panel_async(invE, kp + 2, pbaseA, tid);
    panel_compute(s_panelB, difW, kp + 1, n, hf, koff, mah);
    wait_async0();
    __syncthreads();
  }

  // ---- butterfly-reduce the 16 partials (over N) within each 16-lane half
  #pragma unroll
  for (int v = 0; v < 8; ++v) {
    float p = mah[v];
    p += __shfl_xor(p, 1);
    p += __shfl_xor(p, 2);
    p += __shfl_xor(p, 4);
    p += __shfl_xor(p, 8);
    mah[v] = p;
  }
  if (n == 0) {
    #pragma unroll
    for (int v = 0; v < 8; ++v) {
      const int m = v + 8 * hf;
      out[(size_t)(b0 + m) * kE + e] = mah[v];
    }
  }
}

// ---------------------------------------------------------------------------
extern "C" void kernel_launch(void* const* d_in, const int* in_sizes, int n_in,
                              void* d_out, int out_size, void* d_ws, size_t ws_size,
                              hipStream_t stream) {
  const float* x    = (const float*)d_in[0];   // [B, D]
  const float* cent = (const float*)d_in[1];   // [E, 1, D]
  const float* sig  = (const float*)d_in[2];   // [E, D, D]
  float*       out  = (float*)d_out;           // [B, E]
  float*       inv  = (float*)d_ws;            // needs E*D*D*4 = 8 MB scratch

  invert_sigma_kernel<<<dim3(kE), dim3(256), 0, stream>>>(sig, inv);
  mahal_wmma_kernel<<<dim3(kB / kRowsPerBlk, kE), dim3(128), 0, stream>>>(
      x, cent, inv, out);
}